// Polo_AttentionTransNet_37443524886911
// MI455X (gfx1250) — compile-verified
//
#include <hip/hip_runtime.h>
#include <hip/hip_bf16.h>

typedef __attribute__((ext_vector_type(2))) float v2f;
typedef __attribute__((ext_vector_type(8))) float v8f;

#define BATCH 256

__device__ __forceinline__ v8f wmma_acc(v2f a, v2f b, v8f c) {
  // D = A(16x4 f32) * B(4x16 f32) + C, full fp32 accumulate (matches reference)
  return __builtin_amdgcn_wmma_f32_16x16x4_f32(false, a, false, b, (short)0, c,
                                               false, false);
}

// ---------------------------------------------------------------------------
// Implicit-GEMM 5x5 conv (pad 2, stride 1) + bias + ReLU.
// One wave computes a 16(Cout) x 16(spatial) tile of one image.
// K-reduction reordered as (tap outer, ci inner-by-4): A/B only need to agree
// on K order, so per-tap ih/iw + padding mask are hoisted out of the WMMA loop
// and no integer division is ever needed. Branch-free inner loop: clamped
// addresses + selects instead of exec-masked loads.
// ---------------------------------------------------------------------------
template <int CIN, int H, int W>
__global__ __launch_bounds__(128) void conv5_wmma_kernel(
    const float* __restrict__ x, const float* __restrict__ w,
    const float* __restrict__ bias, float* __restrict__ out, int Cout)
{
  constexpr int K  = CIN * 25;
  constexpr int HW = H * W;

  const int lane = threadIdx.x & 31;
  const int wave = threadIdx.x >> 5;
  const int half = lane >> 4;     // lane group 0/1 -> K slots {0,1} / {2,3}
  const int i16  = lane & 15;

  const int n_img   = blockIdx.z;
  const int m_base  = blockIdx.y << 4;
  const int sp_base = ((blockIdx.x << 2) + wave) << 4;

  const int m  = m_base + i16;    // Cout row this lane feeds into A
  const int hw = sp_base + i16;   // spatial col this lane feeds into B
  const int oh = hw / W;          // W is 16/32 -> shift
  const int ow = hw % W;          // -> mask

  // Clamp instead of zero: rows m>=Cout only affect D rows we never store.
  const int mc = (m < Cout) ? m : 0;
  const float* __restrict__ xn = x + (size_t)n_img * CIN * HW;
  const float* __restrict__ wr = w + (size_t)mc * K;

  v8f acc = {};
  for (int kh = 0; kh < 5; ++kh) {   // rolled: keeps code size I$-friendly
    const int ih = oh + kh - 2;
#pragma unroll
    for (int kw = 0; kw < 5; ++kw) {
      const int iw  = ow + kw - 2;
      const bool vsp = ((unsigned)ih < (unsigned)H) & ((unsigned)iw < (unsigned)W);
      // Clamped per-tap base pointer (offset 0 when padding) + select-to-zero.
      const float* __restrict__ xt = xn + (vsp ? (ih * W + iw) : 0);
      const int tap = kh * 5 + kw;
#pragma unroll
      for (int c0 = 0; c0 < CIN; c0 += 4) {
        v2f a, b;
#pragma unroll
        for (int v = 0; v < 2; ++v) {
          const int ci = c0 + (half << 1) + v;
          const int cs = (ci < CIN) ? ci : 0;      // tail chunk only (CIN=50)
          a[v] = wr[cs * 25 + tap];
          const float t = xt[(size_t)cs * HW];
          b[v] = (vsp && (ci < CIN)) ? t : 0.f;    // conv zero-padding
        }
        acc = wmma_acc(a, b, acc);                 // EXEC all-ones here
      }
    }
  }

#pragma unroll
  for (int v = 0; v < 8; ++v) {
    const int mo = m_base + v + (half << 3);       // D row = v + 8*(lane/16)
    if (mo < Cout) {
      const float val = acc[v] + bias[mo];
      out[(((size_t)n_img * Cout + mo) * H + oh) * W + ow] = fmaxf(val, 0.f);
    }
  }
}

// ---------------------------------------------------------------------------
// GEMM: out[n,m] = act( sum_k feat[n,k] * w[m,k] + bias[m] )
// feat (N,K) row-major, w (M,K) row-major. K compile-time, K % 32 == 0:
// exact unroll-8 trip count -> no remainder path in the binary. Each lane's
// two K-slots are contiguous and 8B-aligned -> v2f (b64) loads, no guards.
// ---------------------------------------------------------------------------
template <int K>
__global__ __launch_bounds__(128) void fc_wmma_kernel(
    const float* __restrict__ feat, const float* __restrict__ w,
    const float* __restrict__ bias, float* __restrict__ out,
    int M, int N, int do_relu)
{
  const int lane = threadIdx.x & 31;
  const int wave = threadIdx.x >> 5;
  const int half = lane >> 4;
  const int i16  = lane & 15;

  const int m_base = blockIdx.y << 4;
  const int n_base = ((blockIdx.x << 2) + wave) << 4;
  const int m = m_base + i16;
  const int n = n_base + i16;
  const int mc = (m < M) ? m : 0;       // clamped rows only feed unsaved D rows
  const int nc = (n < N) ? n : 0;

  const float* __restrict__ wr = w    + (size_t)mc * K + (half << 1);
  const float* __restrict__ fr = feat + (size_t)nc * K + (half << 1);

  v8f acc = {};
#pragma unroll 8
  for (int k0 = 0; k0 < K; k0 += 4) {
    const v2f a = *(const v2f*)(wr + k0);
    const v2f b = *(const v2f*)(fr + k0);
    acc = wmma_acc(a, b, acc);
  }

  if (n < N) {
#pragma unroll
    for (int v = 0; v < 8; ++v) {
      const int mo = m_base + v + (half << 3);
      if (mo < M) {
        float val = acc[v] + bias[mo];
        if (do_relu) val = fmaxf(val, 0.f);
        out[(size_t)n * M + mo] = val;
      }
    }
  }
}

// ---------------------------------------------------------------------------
// 2x2 max pool, writing into a destination with Cdst channels at c_ofs
// (builds the (B,150,2,8) concat buffer directly -> flatten order matches).
// ---------------------------------------------------------------------------
__global__ void maxpool2_kernel(const float* __restrict__ src,
                                float* __restrict__ dst,
                                int C, int H, int W, int Cdst, int c_ofs,
                                int total)
{
  int idx = blockIdx.x * blockDim.x + threadIdx.x;
  if (idx >= total) return;
  const int OW = W >> 1, OH = H >> 1;
  int ow = idx % OW; int t = idx / OW;
  int oh = t % OH;   t /= OH;
  int c  = t % C;    int n = t / C;
  const float* s = src + (((size_t)n * C + c) * H + (oh << 1)) * W + (ow << 1);
  const float v0 = fmaxf(s[0], s[1]);
  const float v1 = fmaxf(s[W], s[W + 1]);
  dst[(((size_t)n * Cdst + (c_ofs + c)) * OH + oh) * OW + ow] = fmaxf(v0, v1);
}

// ---------------------------------------------------------------------------
// fc2 (500 -> 2) + theta identity + z zeros. One block per image, 2 waves:
// wave m does a lane-parallel dot over K=500 and shuffle-reduces.
// ---------------------------------------------------------------------------
__global__ __launch_bounds__(64) void head_kernel(
    const float* __restrict__ h, const float* __restrict__ w2,
    const float* __restrict__ b2, float* __restrict__ out)
{
  const int n    = blockIdx.x;
  const int m    = threadIdx.x >> 5;
  const int lane = threadIdx.x & 31;

  const float* hr = h  + (size_t)n * 500;
  const float* wr = w2 + (size_t)m * 500;
  float s = 0.f;
  for (int k = lane; k < 500; k += 32) s += hr[k] * wr[k];
#pragma unroll
  for (int ofs = 16; ofs > 0; ofs >>= 1) s += __shfl_down(s, ofs, 32);
  if (lane == 0) out[n * 2 + m] = s + b2[m];

  float* theta = out + BATCH * 2;         // (256,2,3) identity
  float* z     = theta + BATCH * 6;       // (256,2) zeros
  if (threadIdx.x < 6)
    theta[n * 6 + threadIdx.x] = (threadIdx.x == 0 || threadIdx.x == 4) ? 1.f : 0.f;
  if (threadIdx.x < 2)
    z[n * 2 + threadIdx.x] = 0.f;
}

// ---------------------------------------------------------------------------
extern "C" void kernel_launch(void* const* d_in, const int* in_sizes, int n_in,
                              void* d_out, int out_size, void* d_ws, size_t ws_size,
                              hipStream_t stream)
{
  (void)in_sizes; (void)n_in; (void)out_size; (void)ws_size;
  // setup_inputs order: 0:x(unused) 1:x_polo_in 2:x_polo_out 3/4:conv1 5/6:conv2a
  // 7/8:conv2b 9/10:fc1 11/12:fc2
  const float* x_in  = (const float*)d_in[1];   // (256,24,4,16)
  const float* x_out = (const float*)d_in[2];   // (256,24,8,32)
  const float* c1w  = (const float*)d_in[3];
  const float* c1b  = (const float*)d_in[4];
  const float* c2aw = (const float*)d_in[5];
  const float* c2ab = (const float*)d_in[6];
  const float* c2bw = (const float*)d_in[7];
  const float* c2bb = (const float*)d_in[8];
  const float* f1w  = (const float*)d_in[9];
  const float* f1b  = (const float*)d_in[10];
  const float* f2w  = (const float*)d_in[11];
  const float* f2b  = (const float*)d_in[12];

  float* ws  = (float*)d_ws;
  float* big = ws;                               // reused pre-pool buf: <= 3,276,800 f
  float* p1  = ws + 3276800;                     // pooled1 (B,50,4,16): 819,200 f
  float* cat = ws + 3276800 + 819200;            // concat (B,150,2,8): 614,400 f
  float* h1  = cat + 614400;                     // fc1 out (B,500): 128,000 f

  const dim3 blk(128);

  // conv1: (B,24,8,32) -> big (B,50,8,32); N=256 spatial -> 16 tiles / 4 waves
  conv5_wmma_kernel<24, 8, 32><<<dim3(4, 4, BATCH), blk, 0, stream>>>(
      x_out, c1w, c1b, big, 50);
  // pool -> p1 (B,50,4,16)
  maxpool2_kernel<<<(819200 + 255) / 256, 256, 0, stream>>>(big, p1, 50, 8, 32,
                                                            50, 0, 819200);
  // conv2a: p1 -> big (B,100,4,16); N=64 -> 4 tiles; M=100 -> 7 tiles
  conv5_wmma_kernel<50, 4, 16><<<dim3(1, 7, BATCH), blk, 0, stream>>>(
      p1, c2aw, c2ab, big, 100);
  // pool -> cat channels [0,100)
  maxpool2_kernel<<<(409600 + 255) / 256, 256, 0, stream>>>(big, cat, 100, 4, 16,
                                                            150, 0, 409600);
  // conv2b: x_in -> big (B,50,4,16)
  conv5_wmma_kernel<24, 4, 16><<<dim3(1, 4, BATCH), blk, 0, stream>>>(
      x_in, c2bw, c2bb, big, 50);
  // pool -> cat channels [100,150)
  maxpool2_kernel<<<(204800 + 255) / 256, 256, 0, stream>>>(big, cat, 50, 4, 16,
                                                            150, 100, 204800);
  // fc1: cat (256,2400) @ f1w(500,2400)^T -> h1 (256,500), ReLU
  fc_wmma_kernel<2400><<<dim3(4, 32), blk, 0, stream>>>(cat, f1w, f1b, h1,
                                                        500, 256, 1);
  // fc2 + theta + z -> d_out [y:512 | theta:1536 | z:512]
  head_kernel<<<dim3(BATCH), dim3(64), 0, stream>>>(h1, f2w, f2b, (float*)d_out);
}